// Decoder_52166672777352
// MI455X (gfx1250) — compile-verified
//
#include <hip/hip_runtime.h>
#include <hip/hip_bf16.h>

#define EPSV 1e-5f

typedef __attribute__((ext_vector_type(2))) float v2f;
typedef __attribute__((ext_vector_type(8))) float v8f;

__device__ __forceinline__ v8f wmma_f32_k4(v2f a, v2f b, v8f c) {
  // V_WMMA_F32_16X16X4_F32 : D = A(16x4) * B(4x16) + C(16x16), all fp32
  return __builtin_amdgcn_wmma_f32_16x16x4_f32(
      /*neg_a=*/false, a, /*neg_b=*/false, b,
      /*c_mod=*/(short)0, c, /*reuse_a=*/false, /*reuse_b=*/false);
}

// ---------------------------------------------------------------------------
// FAST 1x1 conv GEMM (no guards): requires M%32==0, Cin%16==0, Oc%16==0.
// One wave computes a 32(M) x 16(O) tile: 2 accumulators share the B fragment.
// K unrolled by 16 -> 8 WMMAs per iteration. 32-bit offsets -> SADDR loads.
// Y[n,o,t,v] = sum_c W[o,c] * X[n,c,ts,v] + B[o], ts = t/2 if up2.
// ---------------------------------------------------------------------------
__global__ __launch_bounds__(256)
void gemm1x1_wmma_fast(const float* __restrict__ X, const float* __restrict__ W,
                       const float* __restrict__ B, float* __restrict__ Y,
                       int Nn, int Cin, int Tin, int Vv, int Oc, int Tout, int up2)
{
  const int lane = threadIdx.x & 31;
  const int wid  = blockIdx.x * 8 + (threadIdx.x >> 5);
  const int tilesO = Oc >> 4;
  const int M = Nn * Tout * Vv;
  const int tilesM = M >> 5;              // 32-row tiles
  if (wid >= tilesM * tilesO) return;     // wave-uniform exit
  const int tm = wid / tilesO;
  const int to = wid - tm * tilesO;

  const int half = lane >> 4;
  const int lm   = lane & 15;
  const int TVo  = Tout * Vv;
  const unsigned S = (unsigned)(Tin * Vv);   // channel stride in X (elements)

  unsigned aoff0, aoff1;
  {
    const int m0 = tm * 32 + lm;
    const int n0 = m0 / TVo; const int r0 = m0 - n0 * TVo;
    const int t0 = r0 / Vv;  const int v0 = r0 - t0 * Vv;
    aoff0 = (unsigned)n0 * (unsigned)Cin * S +
            (unsigned)(up2 ? (t0 >> 1) : t0) * (unsigned)Vv + (unsigned)v0;
    const int m1 = m0 + 16;
    const int n1 = m1 / TVo; const int r1 = m1 - n1 * TVo;
    const int t1 = r1 / Vv;  const int v1 = r1 - t1 * Vv;
    aoff1 = (unsigned)n1 * (unsigned)Cin * S +
            (unsigned)(up2 ? (t1 >> 1) : t1) * (unsigned)Vv + (unsigned)v1;
  }
  const int o = to * 16 + lm;
  const float2* Wrow = (const float2*)(W + (long long)o * Cin);

  v8f acc0 = {}, acc1 = {};
  for (int c0 = 0; c0 < Cin; c0 += 16) {
    v2f a0[4], a1[4], bf[4];
#pragma unroll
    for (int kk = 0; kk < 4; ++kk) {
      const unsigned k = (unsigned)(c0 + kk * 4 + 2 * half);
      a0[kk].x = X[aoff0 + k * S];
      a0[kk].y = X[aoff0 + (k + 1u) * S];
      a1[kk].x = X[aoff1 + k * S];
      a1[kk].y = X[aoff1 + (k + 1u) * S];
      const float2 w2 = Wrow[k >> 1];
      bf[kk].x = w2.x; bf[kk].y = w2.y;
    }
#pragma unroll
    for (int kk = 0; kk < 4; ++kk) {
      acc0 = wmma_f32_k4(a0[kk], bf[kk], acc0);
      acc1 = wmma_f32_k4(a1[kk], bf[kk], acc1);
    }
  }
  const float bb = B[o];
#pragma unroll
  for (int ms = 0; ms < 2; ++ms) {
    const v8f acc = ms ? acc1 : acc0;
    const int rbase = tm * 32 + ms * 16 + half * 8;
    // decompose once, then carry-increment (rows are consecutive)
    int n2 = rbase / TVo; int r2 = rbase - n2 * TVo;
    int t2 = r2 / Vv;     int v2 = r2 - t2 * Vv;
#pragma unroll
    for (int r = 0; r < 8; ++r) {
      const unsigned yidx =
          (((unsigned)n2 * (unsigned)Oc + (unsigned)o) * (unsigned)Tout +
           (unsigned)t2) * (unsigned)Vv + (unsigned)v2;
      Y[yidx] = acc[r] + bb;
      if (++v2 == Vv) { v2 = 0; if (++t2 == Tout) { t2 = 0; ++n2; } }
    }
  }
}

// ---------------------------------------------------------------------------
// GUARDED 1x1 conv GEMM: for Oc not multiple of 16 (block 9, Oc=9). 16x16 tile.
// ---------------------------------------------------------------------------
__global__ __launch_bounds__(256)
void gemm1x1_wmma_guard(const float* __restrict__ X, const float* __restrict__ W,
                        const float* __restrict__ B, float* __restrict__ Y,
                        int Nn, int Cin, int Tin, int Vv, int Oc, int Tout, int up2)
{
  const int lane = threadIdx.x & 31;
  const int wid  = blockIdx.x * 8 + (threadIdx.x >> 5);
  const int tilesO = (Oc + 15) >> 4;
  const int M = Nn * Tout * Vv;
  const int tilesM = M >> 4;
  if (wid >= tilesM * tilesO) return;
  const int tm = wid / tilesO;
  const int to = wid - tm * tilesO;
  const int half = lane >> 4, lm = lane & 15;
  const int TVo = Tout * Vv;
  const int m = tm * 16 + lm;
  const int n = m / TVo; const int rem = m - n * TVo;
  const int t = rem / Vv; const int v = rem - t * Vv;
  const int ts = up2 ? (t >> 1) : t;
  const unsigned S = (unsigned)(Tin * Vv);
  const unsigned abase = (unsigned)n * (unsigned)Cin * S +
                         (unsigned)ts * (unsigned)Vv + (unsigned)v;
  const int o = to * 16 + lm;
  const bool ov = (o < Oc);

  v8f acc = {};
  for (int c0 = 0; c0 < Cin; c0 += 4) {
    const int k = c0 + 2 * half;
    v2f a, b;
    a.x = (k     < Cin) ? X[abase + (unsigned)k * S]        : 0.0f;
    a.y = (k + 1 < Cin) ? X[abase + (unsigned)(k + 1) * S]  : 0.0f;
    b.x = (ov && k     < Cin) ? W[(long long)o * Cin + k]     : 0.0f;
    b.y = (ov && k + 1 < Cin) ? W[(long long)o * Cin + k + 1] : 0.0f;
    acc = wmma_f32_k4(a, b, acc);
  }
  if (!ov) return;
  const float bb = B[o];
  const int rbase = tm * 16 + half * 8;
  int n2 = rbase / TVo; int r2 = rbase - n2 * TVo;
  int t2 = r2 / Vv;     int v2 = r2 - t2 * Vv;
#pragma unroll
  for (int r = 0; r < 8; ++r) {
    const unsigned yidx =
        (((unsigned)n2 * (unsigned)Oc + (unsigned)o) * (unsigned)Tout +
         (unsigned)t2) * (unsigned)Vv + (unsigned)v2;
    Y[yidx] = acc[r] + bb;
    if (++v2 == Vv) { v2 = 0; if (++t2 == Tout) { t2 = 0; ++n2; } }
  }
}

// ---------------------------------------------------------------------------
// FAST temporal conv GEMM. Z is time-PADDED: (N, C, T+8, V) with 4 zero frames
// each side -> unconditional A loads. Wt pre-transposed (dt, o, c) -> float2 B
// loads. Requires M%32==0, C%16==0.
// ---------------------------------------------------------------------------
__global__ __launch_bounds__(256)
void tcn_gemm_wmma_fast(const float* __restrict__ Z, const float* __restrict__ Wt,
                        const float* __restrict__ B, float* __restrict__ Y,
                        int Nn, int C, int T, int Vv)
{
  const int lane = threadIdx.x & 31;
  const int wid  = blockIdx.x * 8 + (threadIdx.x >> 5);
  const int tilesO = C >> 4;
  const int M = Nn * T * Vv;
  const int tilesM = M >> 5;
  if (wid >= tilesM * tilesO) return;
  const int tm = wid / tilesO;
  const int to = wid - tm * tilesO;
  const int half = lane >> 4, lm = lane & 15;
  const int TV = T * Vv;
  const unsigned Sp = (unsigned)((T + 8) * Vv);   // padded channel stride

  unsigned aoff0, aoff1;
  {
    const int m0 = tm * 32 + lm;
    const int n0 = m0 / TV; const int r0 = m0 - n0 * TV;
    const int t0 = r0 / Vv; const int v0 = r0 - t0 * Vv;
    aoff0 = (unsigned)n0 * (unsigned)C * Sp + (unsigned)t0 * (unsigned)Vv + (unsigned)v0;
    const int m1 = m0 + 16;
    const int n1 = m1 / TV; const int r1 = m1 - n1 * TV;
    const int t1 = r1 / Vv; const int v1 = r1 - t1 * Vv;
    aoff1 = (unsigned)n1 * (unsigned)C * Sp + (unsigned)t1 * (unsigned)Vv + (unsigned)v1;
  }
  const int o = to * 16 + lm;

  v8f acc0 = {}, acc1 = {};
  for (int dt = 0; dt < 9; ++dt) {
    const float2* Wr = (const float2*)(Wt + ((long long)dt * C + o) * C);
    const unsigned doff = (unsigned)(dt * Vv);
    for (int c0 = 0; c0 < C; c0 += 16) {
      v2f a0[4], a1[4], bf[4];
#pragma unroll
      for (int kk = 0; kk < 4; ++kk) {
        const unsigned k = (unsigned)(c0 + kk * 4 + 2 * half);
        a0[kk].x = Z[aoff0 + doff + k * Sp];
        a0[kk].y = Z[aoff0 + doff + (k + 1u) * Sp];
        a1[kk].x = Z[aoff1 + doff + k * Sp];
        a1[kk].y = Z[aoff1 + doff + (k + 1u) * Sp];
        const float2 w2 = Wr[k >> 1];
        bf[kk].x = w2.x; bf[kk].y = w2.y;
      }
#pragma unroll
      for (int kk = 0; kk < 4; ++kk) {
        acc0 = wmma_f32_k4(a0[kk], bf[kk], acc0);
        acc1 = wmma_f32_k4(a1[kk], bf[kk], acc1);
      }
    }
  }
  const float bb = B[o];
#pragma unroll
  for (int ms = 0; ms < 2; ++ms) {
    const v8f acc = ms ? acc1 : acc0;
    const int rbase = tm * 32 + ms * 16 + half * 8;
    int n2 = rbase / TV; int r2 = rbase - n2 * TV;
    int t2 = r2 / Vv;    int v2 = r2 - t2 * Vv;
#pragma unroll
    for (int r = 0; r < 8; ++r) {
      const unsigned yidx =
          (((unsigned)n2 * (unsigned)C + (unsigned)o) * (unsigned)T +
           (unsigned)t2) * (unsigned)Vv + (unsigned)v2;
      Y[yidx] = acc[r] + bb;
      if (++v2 == Vv) { v2 = 0; if (++t2 == T) { t2 = 0; ++n2; } }
    }
  }
}

// GUARDED temporal conv GEMM (block 9, C=3). Padded Z, transposed Wt.
__global__ __launch_bounds__(256)
void tcn_gemm_wmma_guard(const float* __restrict__ Z, const float* __restrict__ Wt,
                         const float* __restrict__ B, float* __restrict__ Y,
                         int Nn, int C, int T, int Vv)
{
  const int lane = threadIdx.x & 31;
  const int wid  = blockIdx.x * 8 + (threadIdx.x >> 5);
  const int tilesO = (C + 15) >> 4;
  const int M = Nn * T * Vv;
  const int tilesM = M >> 4;
  if (wid >= tilesM * tilesO) return;
  const int tm = wid / tilesO;
  const int to = wid - tm * tilesO;
  const int half = lane >> 4, lm = lane & 15;
  const int TV = T * Vv;
  const unsigned Sp = (unsigned)((T + 8) * Vv);
  const int m = tm * 16 + lm;
  const int n = m / TV; const int rem = m - n * TV;
  const int t = rem / Vv; const int v = rem - t * Vv;
  const unsigned abase = (unsigned)n * (unsigned)C * Sp +
                         (unsigned)t * (unsigned)Vv + (unsigned)v;
  const int o = to * 16 + lm;
  const bool ov = (o < C);

  v8f acc = {};
  for (int dt = 0; dt < 9; ++dt) {
    const unsigned doff = (unsigned)(dt * Vv);
    for (int c0 = 0; c0 < C; c0 += 4) {
      const int k = c0 + 2 * half;
      v2f a, b;
      a.x = (k     < C) ? Z[abase + doff + (unsigned)k * Sp]       : 0.0f;
      a.y = (k + 1 < C) ? Z[abase + doff + (unsigned)(k + 1) * Sp] : 0.0f;
      b.x = (ov && k     < C) ? Wt[((long long)dt * C + o) * C + k]     : 0.0f;
      b.y = (ov && k + 1 < C) ? Wt[((long long)dt * C + o) * C + k + 1] : 0.0f;
      acc = wmma_f32_k4(a, b, acc);
    }
  }
  if (!ov) return;
  const float bb = B[o];
  const int rbase = tm * 16 + half * 8;
  int n2 = rbase / TV; int r2 = rbase - n2 * TV;
  int t2 = r2 / Vv;    int v2 = r2 - t2 * Vv;
#pragma unroll
  for (int r = 0; r < 8; ++r) {
    const unsigned yidx =
        (((unsigned)n2 * (unsigned)C + (unsigned)o) * (unsigned)T +
         (unsigned)t2) * (unsigned)Vv + (unsigned)v2;
    Y[yidx] = acc[r] + bb;
    if (++v2 == Vv) { v2 = 0; if (++t2 == T) { t2 = 0; ++n2; } }
  }
}

// Wt[dt*C*C + o*C + c] = W[(o*C + c)*9 + dt]
__global__ void tcn_transpose_k(const float* __restrict__ W, float* __restrict__ Wt,
                                int C, int total)
{
  for (int idx = blockIdx.x * blockDim.x + threadIdx.x; idx < total;
       idx += blockDim.x * gridDim.x) {
    const int c  = idx % C;
    const int o  = (idx / C) % C;
    const int dt = idx / (C * C);
    Wt[idx] = W[((long long)o * C + c) * 9 + dt];
  }
}

// zero the 4 leading + 4 trailing time frames of padded (N,C,Tp,V) buffer
__global__ void zero_pad_k(float* __restrict__ Z, int C, int Tp, int total)
{
  const int Vv = 25;
  for (int idx = blockIdx.x * blockDim.x + threadIdx.x; idx < total;
       idx += blockDim.x * gridDim.x) {
    const int v = idx % Vv;
    const int s = (idx / Vv) % 8;
    const int c = (idx / (Vv * 8)) % C;
    const int n = idx / (Vv * 8 * C);
    const int t = (s < 4) ? s : (Tp - 8 + s);
    Z[(((unsigned)n * C + c) * (unsigned)Tp + (unsigned)t) * Vv + v] = 0.0f;
  }
}

// ---------------------------------------------------------------------------
// Graph aggregation: Y2[n,c,t,w] = sum_k sum_v Y1[n,k*C+c,t,v] * (A*imp)[k,v,w]
// ---------------------------------------------------------------------------
__global__ __launch_bounds__(256)
void aggregate_k(const float* __restrict__ Y1, const float* __restrict__ Aadj,
                 const float* __restrict__ Eimp, float* __restrict__ Y2,
                 int Nn, int Cout, int T, int total)
{
  __shared__ float aimp[3 * 25 * 25];
  for (int i = threadIdx.x; i < 3 * 25 * 25; i += blockDim.x)
    aimp[i] = Aadj[i] * Eimp[i];
  __syncthreads();
  const int Vv = 25;
  for (int idx = blockIdx.x * blockDim.x + threadIdx.x; idx < total;
       idx += blockDim.x * gridDim.x) {
    const int w = idx % Vv;
    const int t = (idx / Vv) % T;
    const int c = (idx / (Vv * T)) % Cout;
    const int n = idx / (Vv * T * Cout);
    float accv = 0.0f;
#pragma unroll
    for (int k = 0; k < 3; ++k) {
      const float* yp = Y1 + (((unsigned)n * 3u * Cout + (unsigned)k * Cout + c) *
                              (unsigned)T + t) * (unsigned)Vv;
      const float* ap = aimp + k * 25 * 25;
#pragma unroll
      for (int v = 0; v < 25; ++v) accv += yp[v] * ap[v * 25 + w];
    }
    Y2[idx] = accv;
  }
}

// ---------------------------------------------------------------------------
// BN helpers (training-mode stats over N,T,V per channel)
// ---------------------------------------------------------------------------
__global__ __launch_bounds__(256)
void bn_stats_k(const float* __restrict__ X, float* __restrict__ stats,
                int C, int TV, int per)
{
  __shared__ float ss[256], ss2[256];
  const int c = blockIdx.x;
  float s = 0.0f, s2 = 0.0f;
  for (int i = blockIdx.y * blockDim.x + threadIdx.x; i < per;
       i += blockDim.x * gridDim.y) {
    const int n = i / TV;
    const int j = i - n * TV;
    const float val = X[((unsigned)n * C + c) * (unsigned)TV + (unsigned)j];
    s += val; s2 += val * val;
  }
  ss[threadIdx.x] = s; ss2[threadIdx.x] = s2;
  __syncthreads();
  for (int st = 128; st > 0; st >>= 1) {
    if ((int)threadIdx.x < st) {
      ss[threadIdx.x]  += ss[threadIdx.x + st];
      ss2[threadIdx.x] += ss2[threadIdx.x + st];
    }
    __syncthreads();
  }
  if (threadIdx.x == 0) {
    atomicAdd(&stats[c], ss[0]);
    atomicAdd(&stats[C + c], ss2[0]);
  }
}

__global__ void zero_k(float* __restrict__ p, int n) {
  const int i = blockIdx.x * blockDim.x + threadIdx.x;
  if (i < n) p[i] = 0.0f;
}

// BN -> ReLU -> optional x2 upsample, writing into PADDED (N,C,Tout+8,V) buffer
__global__ __launch_bounds__(256)
void bn_relu_up_pad_k(const float* __restrict__ X, const float* __restrict__ stats,
                      const float* __restrict__ g, const float* __restrict__ b,
                      float* __restrict__ Y, int Nn, int C, int Tin, int up2,
                      float invCnt, int total)
{
  const int Vv = 25;
  const int Tout = up2 ? 2 * Tin : Tin;
  const int Tp = Tout + 8;
  for (int idx = blockIdx.x * blockDim.x + threadIdx.x; idx < total;
       idx += blockDim.x * gridDim.x) {
    const int v = idx % Vv;
    const int t = (idx / Vv) % Tout;
    const int c = (idx / (Vv * Tout)) % C;
    const int n = idx / (Vv * Tout * C);
    const int ts = up2 ? (t >> 1) : t;
    const float mean = stats[c] * invCnt;
    float var = stats[C + c] * invCnt - mean * mean;
    var = fmaxf(var, 0.0f);
    const float sc = rsqrtf(var + EPSV) * g[c];
    const float val =
        (X[(((unsigned)n * C + c) * (unsigned)Tin + (unsigned)ts) * Vv + v] - mean) * sc + b[c];
    Y[(((unsigned)n * C + c) * (unsigned)Tp + (unsigned)(t + 4)) * Vv + v] = fmaxf(val, 0.0f);
  }
}

// Epilogue: ReLU( BN2(tcn) + residual ), residual = none / identity / BN3(proj)
__global__ __launch_bounds__(256)
void final_k(const float* __restrict__ Tc, const float* __restrict__ st2,
             const float* __restrict__ g2, const float* __restrict__ b2,
             const float* __restrict__ Res, const float* __restrict__ st3,
             const float* __restrict__ g3, const float* __restrict__ b3,
             int resMode, float* __restrict__ Out, int C, int T,
             float invCnt, int total)
{
  const int Vv = 25;
  for (int idx = blockIdx.x * blockDim.x + threadIdx.x; idx < total;
       idx += blockDim.x * gridDim.x) {
    const int c = (idx / (Vv * T)) % C;
    const float mean2 = st2[c] * invCnt;
    float var2 = st2[C + c] * invCnt - mean2 * mean2;
    var2 = fmaxf(var2, 0.0f);
    float val = (Tc[idx] - mean2) * rsqrtf(var2 + EPSV) * g2[c] + b2[c];
    float r = 0.0f;
    if (resMode == 1) {
      r = Res[idx];
    } else if (resMode == 2) {
      const float mean3 = st3[c] * invCnt;
      float var3 = st3[C + c] * invCnt - mean3 * mean3;
      var3 = fmaxf(var3, 0.0f);
      r = (Res[idx] - mean3) * rsqrtf(var3 + EPSV) * g3[c] + b3[c];
    }
    Out[idx] = fmaxf(val + r, 0.0f);
  }
}

// ---------------------------------------------------------------------------
extern "C" void kernel_launch(void* const* d_in, const int* in_sizes, int n_in,
                              void* d_out, int out_size, void* d_ws, size_t ws_size,
                              hipStream_t stream)
{
  (void)in_sizes; (void)n_in; (void)out_size; (void)ws_size;
  static const int CFGi[10][4] = {
      {256, 256, 1, 1}, {256, 256, 1, 1}, {256, 128, 2, 1},
      {128, 128, 1, 1}, {128, 128, 1, 1}, {128, 64, 2, 1},
      {64, 64, 1, 1},   {64, 64, 1, 1},   {64, 64, 1, 1},
      {64, 3, 1, 0}};
  const int Nn = 32, Vv = 25;

  const float* x    = (const float*)d_in[0];
  const float* Aadj = (const float*)d_in[1];
  const float* Eimp = (const float*)d_in[2];
  int cur_idx = 3;

  char* ws = (char*)d_ws;
  const size_t ACT  = 15360000ull * 4ull;   // activation ping-pong
  const size_t GSZ  = 46080000ull * 4ull;   // GCN out (also residual proj)
  const size_t PACT = 68000000ull;          // padded TCN input (max 16.99M f32)
  const size_t WTSZ = 2359296ull;           // transposed tcn weights (256*256*9)
  float* bufA  = (float*)(ws);
  float* bufB  = (float*)(ws + ACT);
  float* bufG  = (float*)(ws + 2 * ACT);
  float* bufR  = bufG;                      // lifetime-disjoint with bufG use
  float* bufT1 = (float*)(ws + 2 * ACT + GSZ);
  float* bufT2 = (float*)(ws + 3 * ACT + GSZ);          // padded
  float* bufWT = (float*)(ws + 3 * ACT + GSZ + PACT);
  float* st1   = (float*)(ws + 3 * ACT + GSZ + PACT + WTSZ);
  float* st2   = st1 + 512;
  float* st3   = st1 + 1024;

  const float* cur = x;
  int Tin = 75;
  int flip = 0;

  for (int i = 0; i < 10; ++i) {
    const int cin = CFGi[i][0], cout = CFGi[i][1];
    const int stride = CFGi[i][2], residual = CFGi[i][3];
    const int up2 = (stride == 2);
    const int Tout = up2 ? 2 * Tin : Tin;
    const int Tp = Tout + 8;

    const float* gcn_w = (const float*)d_in[cur_idx++];
    const float* gcn_b = (const float*)d_in[cur_idx++];
    const float* bn1g  = (const float*)d_in[cur_idx++];
    const float* bn1b  = (const float*)d_in[cur_idx++];
    const float* tcn_w = (const float*)d_in[cur_idx++];
    const float* tcn_b = (const float*)d_in[cur_idx++];
    const float* bn2g  = (const float*)d_in[cur_idx++];
    const float* bn2b  = (const float*)d_in[cur_idx++];
    const float *res_w = nullptr, *res_b = nullptr, *res_bng = nullptr, *res_bnb = nullptr;
    const bool resproj = residual && (cin != cout || stride != 1);
    if (resproj) {
      res_w   = (const float*)d_in[cur_idx++];
      res_b   = (const float*)d_in[cur_idx++];
      res_bng = (const float*)d_in[cur_idx++];
      res_bnb = (const float*)d_in[cur_idx++];
    }

    float* outAct = (i == 9) ? (float*)d_out : (flip ? bufB : bufA);

    // 1. GCN 1x1 GEMM -> bufG (O = 3*cout)
    const int O1 = 3 * cout;
    {
      const int M = Nn * Tin * Vv;
      if ((O1 & 15) == 0) {
        const int tiles = (M >> 5) * (O1 >> 4);
        gemm1x1_wmma_fast<<<(tiles + 7) / 8, 256, 0, stream>>>(
            cur, gcn_w, gcn_b, bufG, Nn, cin, Tin, Vv, O1, Tin, 0);
      } else {
        const int tiles = (M >> 4) * ((O1 + 15) >> 4);
        gemm1x1_wmma_guard<<<(tiles + 7) / 8, 256, 0, stream>>>(
            cur, gcn_w, gcn_b, bufG, Nn, cin, Tin, Vv, O1, Tin, 0);
      }
    }
    // 2. adjacency aggregation -> bufT1
    {
      const int total = Nn * cout * Tin * Vv;
      aggregate_k<<<(total + 255) / 256, 256, 0, stream>>>(
          bufG, Aadj, Eimp + (long long)i * 3 * 25 * 25, bufT1, Nn, cout, Tin, total);
    }
    // 3. BN1 stats; BN1+ReLU(+upsample) into padded bufT2; zero pads
    zero_k<<<(2 * cout + 255) / 256, 256, 0, stream>>>(st1, 2 * cout);
    bn_stats_k<<<dim3(cout, 8), 256, 0, stream>>>(bufT1, st1, cout, Tin * Vv, Nn * Tin * Vv);
    {
      const int ptotal = Nn * cout * 8 * Vv;
      zero_pad_k<<<(ptotal + 255) / 256, 256, 0, stream>>>(bufT2, cout, Tp, ptotal);
      const int total = Nn * cout * Tout * Vv;
      bn_relu_up_pad_k<<<(total + 255) / 256, 256, 0, stream>>>(
          bufT1, st1, bn1g, bn1b, bufT2, Nn, cout, Tin, up2,
          1.0f / (float)(Nn * Tin * Vv), total);
    }
    // 4. transpose tcn weights; temporal conv GEMM -> bufT1
    {
      const int wtot = cout * cout * 9;
      tcn_transpose_k<<<(wtot + 255) / 256, 256, 0, stream>>>(tcn_w, bufWT, cout, wtot);
      const int M = Nn * Tout * Vv;
      if ((cout & 15) == 0) {
        const int tiles = (M >> 5) * (cout >> 4);
        tcn_gemm_wmma_fast<<<(tiles + 7) / 8, 256, 0, stream>>>(
            bufT2, bufWT, tcn_b, bufT1, Nn, cout, Tout, Vv);
      } else {
        const int tiles = (M >> 4) * ((cout + 15) >> 4);
        tcn_gemm_wmma_guard<<<(tiles + 7) / 8, 256, 0, stream>>>(
            bufT2, bufWT, tcn_b, bufT1, Nn, cout, Tout, Vv);
      }
    }
    // 5. BN2 stats
    zero_k<<<(2 * cout + 255) / 256, 256, 0, stream>>>(st2, 2 * cout);
    bn_stats_k<<<dim3(cout, 8), 256, 0, stream>>>(bufT1, st2, cout, Tout * Vv, Nn * Tout * Vv);
    // 6. residual branch
    int resMode = 0;
    const float* Res = nullptr;
    if (residual) {
      if (!resproj) {
        resMode = 1; Res = cur;
      } else {
        resMode = 2; Res = bufR;
        const int M = Nn * Tout * Vv;
        const int tiles = (M >> 5) * (cout >> 4);   // cout multiple of 16 here
        gemm1x1_wmma_fast<<<(tiles + 7) / 8, 256, 0, stream>>>(
            cur, res_w, res_b, bufR, Nn, cin, Tin, Vv, cout, Tout, up2);
        zero_k<<<(2 * cout + 255) / 256, 256, 0, stream>>>(st3, 2 * cout);
        bn_stats_k<<<dim3(cout, 8), 256, 0, stream>>>(bufR, st3, cout, Tout * Vv, Nn * Tout * Vv);
      }
    }
    // 7. fused BN2 + residual + ReLU -> next activation
    {
      const int total = Nn * cout * Tout * Vv;
      final_k<<<(total + 255) / 256, 256, 0, stream>>>(
          bufT1, st2, bn2g, bn2b, Res, st3, res_bng, res_bnb, resMode,
          outAct, cout, Tout, 1.0f / (float)(Nn * Tout * Vv), total);
    }

    cur = outAct;
    Tin = Tout;
    flip ^= 1;
  }
}